// LowRankMixtureCrossNet_24481313587242
// MI455X (gfx1250) — compile-verified
//
#include <hip/hip_runtime.h>
#include <hip/hip_bf16.h>

#define B_ 16384
#define N_ 1024
#define L_ 3
#define E_ 8
#define R_ 64
#define ER_ (E_*R_)   // 512

typedef float v2f __attribute__((ext_vector_type(2)));
typedef float v8f __attribute__((ext_vector_type(8)));

// D = A(16x4) * B(4x16) + C, fp32 WMMA
__device__ __forceinline__ v8f wmma_f32(v2f a, v2f b, v8f c) {
    return __builtin_amdgcn_wmma_f32_16x16x4_f32(
        /*neg_a=*/false, a, /*neg_b=*/false, b,
        /*c_mod=*/(short)0, c, /*reuse_a=*/false, /*reuse_b=*/false);
}

__global__ __launch_bounds__(256)
void crossnet_fused_kernel(const float* __restrict__ x,
                           const float* __restrict__ U,     // [L][E][N][R]
                           const float* __restrict__ V,     // [L][E][R][N]
                           const float* __restrict__ C,     // [L][E][R][R]
                           const float* __restrict__ bias,  // [L][N]
                           const float* __restrict__ gate_w,// [E][N]
                           float* __restrict__ out)         // [B][N]
{
    __shared__ float x0_s[16 * N_];    // 64 KB
    __shared__ float xl_s[16 * N_];    // 64 KB
    __shared__ float v_s [16 * ER_];   // 32 KB : relu(v)
    __shared__ float act_s[16 * ER_];  // 32 KB : gates * relu(c)
    __shared__ float gates_s[16 * E_];

    const int tid  = threadIdx.x;
    const int lane = tid & 31;
    const int wave = tid >> 5;          // 0..7
    const int mrow = lane & 15;         // N-index (B-operand / D) or M-index (A-operand)
    const int ksel = lane >> 4;         // half-wave: K pair selector
    const size_t row0 = (size_t)blockIdx.x * 16;

    // ---------- load x tile (resident for all 3 layers) ----------
    {
        const float4* xg  = (const float4*)(x + row0 * N_);
        float4* x0v = (float4*)x0_s;
        float4* xlv = (float4*)xl_s;
        #pragma unroll
        for (int i = 0; i < (16 * N_ / 4) / 256; ++i) {
            int idx = tid + i * 256;
            float4 t = xg[idx];
            x0v[idx] = t;
            xlv[idx] = t;
        }
    }
    __syncthreads();

    for (int l = 0; l < L_; ++l) {
        // ---------- Phase A: gate logits + softmax ----------
        if (tid < 16 * E_) {
            const int row = tid >> 3, e = tid & 7;
            const float* gw = gate_w + e * N_;
            const float* xr = xl_s + row * N_;
            float acc = 0.f;
            for (int n = 0; n < N_; ++n) acc = fmaf(xr[n], gw[n], acc);
            gates_s[tid] = acc;
        }
        __syncthreads();
        if (tid < 16) {
            float m = gates_s[tid * E_];
            #pragma unroll
            for (int e = 1; e < E_; ++e) m = fmaxf(m, gates_s[tid * E_ + e]);
            float tmp[E_]; float s = 0.f;
            #pragma unroll
            for (int e = 0; e < E_; ++e) { tmp[e] = __expf(gates_s[tid * E_ + e] - m); s += tmp[e]; }
            const float inv = 1.f / s;
            #pragma unroll
            for (int e = 0; e < E_; ++e) gates_s[tid * E_ + e] = tmp[e] * inv;
        }
        // ---------- Phase B: v = x_l @ V[l][e]^T  (wave <-> expert) ----------
        {
            const int e = wave;
            const float* Vb = V + ((size_t)l * E_ + e) * (R_ * N_);  // [R][N]
            v8f a0 = {}, a1 = {}, a2 = {}, a3 = {};
            const float* a_src = xl_s + mrow * N_ + ksel * 2;
            for (int n0 = 0; n0 < N_; n0 += 4) {
                v2f a = *(const v2f*)(a_src + n0);
                const float* bp = Vb + n0 + ksel * 2;
                v2f b0 = *(const v2f*)(bp + (0 * 16 + mrow) * N_);
                v2f b1 = *(const v2f*)(bp + (1 * 16 + mrow) * N_);
                v2f b2 = *(const v2f*)(bp + (2 * 16 + mrow) * N_);
                v2f b3 = *(const v2f*)(bp + (3 * 16 + mrow) * N_);
                a0 = wmma_f32(a, b0, a0);
                a1 = wmma_f32(a, b1, a1);
                a2 = wmma_f32(a, b2, a2);
                a3 = wmma_f32(a, b3, a3);
            }
            v8f acc[4] = {a0, a1, a2, a3};
            #pragma unroll
            for (int t = 0; t < 4; ++t)
                #pragma unroll
                for (int j = 0; j < 8; ++j) {
                    const int row = j + 8 * ksel;
                    v_s[row * ER_ + e * R_ + t * 16 + mrow] = fmaxf(acc[t][j], 0.f);
                }
        }
        __syncthreads();

        // ---------- Phase C: c = C[l][e] @ relu(v); store gates*relu(c) ----------
        {
            const int e = wave;
            const float* Cb = C + ((size_t)l * E_ + e) * (R_ * R_);  // [R][S]
            v8f a0 = {}, a1 = {}, a2 = {}, a3 = {};
            const float* a_src = v_s + mrow * ER_ + e * R_ + ksel * 2;
            #pragma unroll
            for (int s0 = 0; s0 < R_; s0 += 4) {
                v2f a = *(const v2f*)(a_src + s0);
                const float* bp = Cb + s0 + ksel * 2;
                v2f b0 = *(const v2f*)(bp + (0 * 16 + mrow) * R_);
                v2f b1 = *(const v2f*)(bp + (1 * 16 + mrow) * R_);
                v2f b2 = *(const v2f*)(bp + (2 * 16 + mrow) * R_);
                v2f b3 = *(const v2f*)(bp + (3 * 16 + mrow) * R_);
                a0 = wmma_f32(a, b0, a0);
                a1 = wmma_f32(a, b1, a1);
                a2 = wmma_f32(a, b2, a2);
                a3 = wmma_f32(a, b3, a3);
            }
            v8f acc[4] = {a0, a1, a2, a3};
            #pragma unroll
            for (int t = 0; t < 4; ++t)
                #pragma unroll
                for (int j = 0; j < 8; ++j) {
                    const int row = j + 8 * ksel;
                    const float g = gates_s[row * E_ + e];
                    act_s[row * ER_ + e * R_ + t * 16 + mrow] = g * fmaxf(acc[t][j], 0.f);
                }
        }
        __syncthreads();

        // ---------- Phase D: gated_u = act @ W  ([16,512]x[512,1024]) ----------
        // W[(e,r), n] = U[l][e][n][r]; wave <-> 128-column slice; fused residual.
        {
            const int n_base = wave * 128;
            const float* Ub = U + (size_t)l * (E_ * N_ * R_);   // [E][N][R]
            v8f acc[8] = {};
            const float* a_src = act_s + mrow * ER_ + ksel * 2;
            for (int k0 = 0; k0 < ER_; k0 += 4) {
                v2f a = *(const v2f*)(a_src + k0);
                const int e  = k0 >> 6;
                const int r0 = (k0 & 63) + ksel * 2;
                const float* bp = Ub + ((size_t)e * N_) * R_ + r0;
                #pragma unroll
                for (int t = 0; t < 8; ++t) {
                    const int n = n_base + t * 16 + mrow;
                    v2f b = *(const v2f*)(bp + (size_t)n * R_);
                    acc[t] = wmma_f32(a, b, acc[t]);
                }
            }
            // x_l[b,n] += x0[b,n] * (gated_u + bias[l][n])   (sum_e gates == 1)
            const float* bl = bias + l * N_;
            #pragma unroll
            for (int t = 0; t < 8; ++t) {
                const int n = n_base + t * 16 + mrow;
                const float bn = bl[n];
                #pragma unroll
                for (int j = 0; j < 8; ++j) {
                    const int row = j + 8 * ksel;
                    const float gu  = acc[t][j];
                    const float x0v = x0_s[row * N_ + n];
                    const float xlv = xl_s[row * N_ + n];
                    xl_s[row * N_ + n] = fmaf(x0v, gu + bn, xlv);
                }
            }
        }
        __syncthreads();
    }

    // ---------- write out ----------
    {
        float4* og = (float4*)(out + row0 * N_);
        const float4* xlv = (const float4*)xl_s;
        #pragma unroll
        for (int i = 0; i < (16 * N_ / 4) / 256; ++i) {
            int idx = tid + i * 256;
            og[idx] = xlv[idx];
        }
    }
}

extern "C" void kernel_launch(void* const* d_in, const int* in_sizes, int n_in,
                              void* d_out, int out_size, void* d_ws, size_t ws_size,
                              hipStream_t stream) {
    const float* x      = (const float*)d_in[0];   // [B,N]
    const float* U      = (const float*)d_in[1];   // [L,E,N,R]
    const float* V      = (const float*)d_in[2];   // [L,E,R,N]
    const float* C      = (const float*)d_in[3];   // [L,E,R,R]
    const float* bias   = (const float*)d_in[4];   // [L,N]
    const float* gate_w = (const float*)d_in[5];   // [E,N]
    float* out = (float*)d_out;

    dim3 grid(B_ / 16);
    dim3 block(256);
    crossnet_fused_kernel<<<grid, block, 0, stream>>>(x, U, V, C, bias, gate_w, out);
}